// DictionaryLearningTokenized_45801531244952
// MI455X (gfx1250) — compile-verified
//
#include <hip/hip_runtime.h>
#include <hip/hip_bf16.h>

#define EMBED_DIM 64
#define NUM_ATOMS 1024
#define SPARSITY  8
#define N_PIXELS  16384
#define DIAG_EPS  0.0001f
#define CHOL_EPS  1e-6f

typedef __attribute__((ext_vector_type(2))) float v2f;
typedef __attribute__((ext_vector_type(8))) float v8f;

// ---------------------------------------------------------------------------
// C[M x 1024] = A^T * B  (+ diag_eps on the diagonal), K = 64.
// A is [64 x lda] (row m of the logical A^T is A[:,m]), B is [64 x ldb].
// One wave computes a 16(M) x 64(N) strip using V_WMMA_F32_16X16X4_F32.
// ---------------------------------------------------------------------------
__global__ __launch_bounds__(128) void wmma_gemm_tn(
    const float* __restrict__ A, int lda,
    const float* __restrict__ B, int ldb,
    float* __restrict__ C, int ldc,
    float diag_eps)
{
    const int lane  = threadIdx.x & 31;
    const int wid   = threadIdx.x >> 5;
    const int wtile = blockIdx.x * 4 + wid;
    const int tilesPerRow = NUM_ATOMS / 64;          // 16 N-strips
    const int mbase = (wtile / tilesPerRow) * 16;
    const int nbase = (wtile % tilesPerRow) * 64;
    const int m     = lane & 15;                      // M row (A) / N col (B)
    const int koff  = (lane >> 4) << 1;               // lanes 0-15: K 0,1 ; 16-31: K 2,3

    v8f acc[4] = {};
    for (int k0 = 0; k0 < EMBED_DIM; k0 += 4) {
        const int ka = k0 + koff;
        v2f a;
        a.x = A[ka * lda + mbase + m];
        a.y = A[(ka + 1) * lda + mbase + m];
#pragma unroll
        for (int j = 0; j < 4; ++j) {
            v2f b;
            b.x = B[ka * ldb + nbase + j * 16 + m];
            b.y = B[(ka + 1) * ldb + nbase + j * 16 + m];
            acc[j] = __builtin_amdgcn_wmma_f32_16x16x4_f32(
                false, a, false, b, (short)0, acc[j], false, false);
        }
    }

    const int half = (lane >> 4) * 8;   // C/D layout: lanes 16-31 hold rows M+8
#pragma unroll
    for (int r = 0; r < 8; ++r) {
        const int row = mbase + r + half;
        float* crow = C + (size_t)row * ldc + nbase + m;
#pragma unroll
        for (int j = 0; j < 4; ++j) {
            float v = acc[j][r];
            if (row == nbase + j * 16 + m) v += diag_eps;
            crow[j * 16] = v;
        }
    }
}

// ---------------------------------------------------------------------------
// Batched OMP: one wave32 per pixel. h / h_bar rows live in 64 VGPRs/lane
// (atom 32t+lane -> slot t). Cholesky / triangular solves are wave-uniform
// and fully unrolled into registers.
// ---------------------------------------------------------------------------
__global__ __launch_bounds__(256) void omp_kernel(
    const float* __restrict__ G,
    const float* __restrict__ hbar,
    const float* __restrict__ Dm,
    float* __restrict__ recon,
    float* __restrict__ outI,
    float* __restrict__ outC)
{
    const int lane = threadIdx.x & 31;
    const int wid  = threadIdx.x >> 5;
    const int n    = blockIdx.x * 8 + wid;

    const float* hb = hbar + (size_t)n * NUM_ATOMS;

    float h[32], hb_r[32];
#pragma unroll
    for (int t = 0; t < 32; ++t) {
        hb_r[t] = hb[t * 32 + lane];
        h[t]    = hb_r[t];
    }
    unsigned mask = 0u;

    int   I_[SPARSITY];
    float L[SPARSITY * (SPARSITY + 1) / 2];   // packed lower triangular
    float hsel[SPARSITY];
    float cf[SPARSITY];

#pragma unroll
    for (int s = 0; s < SPARSITY; ++s) {
        // ---- masked argmax of |h| over 1024 atoms (first-max tie-break) ----
        float bv = -2.0f; int bi = NUM_ATOMS;
#pragma unroll
        for (int t = 0; t < 32; ++t) {
            const float sc = ((mask >> t) & 1u) ? -1.0f : fabsf(h[t]);
            const int   aj = t * 32 + lane;
            if (sc > bv || (sc == bv && aj < bi)) { bv = sc; bi = aj; }
        }
#pragma unroll
        for (int off = 16; off > 0; off >>= 1) {
            const float ov = __shfl_xor(bv, off);
            const int   oi = __shfl_xor(bi, off);
            if (ov > bv || (ov == bv && oi < bi)) { bv = ov; bi = oi; }
        }
        const int idx = bi;                           // wave-uniform
        if ((idx & 31) == lane) mask |= (1u << (idx >> 5));

        const float diag = G[(size_t)idx * NUM_ATOMS + idx];  // includes +eps
        hsel[s] = hb[idx];

        // ---- rank-1 Cholesky update ----
        if (s == 0) {
            L[0] = sqrtf(fmaxf(diag, CHOL_EPS));
        } else {
            float w[SPARSITY];
            float nrm = 0.0f;
#pragma unroll
            for (int i = 0; i < s; ++i) {
                float a = G[(size_t)idx * NUM_ATOMS + I_[i]];   // G symmetric
#pragma unroll
                for (int j = 0; j < i; ++j) a -= L[i * (i + 1) / 2 + j] * w[j];
                w[i] = a / L[i * (i + 1) / 2 + i];
                L[s * (s + 1) / 2 + i] = w[i];
                nrm += w[i] * w[i];
            }
            L[s * (s + 1) / 2 + s] = sqrtf(fmaxf(diag - nrm, CHOL_EPS));
        }
        I_[s] = idx;

        // ---- solve L y = hsel, then L^T cf = y (size s+1) ----
        float y[SPARSITY];
#pragma unroll
        for (int i = 0; i <= s; ++i) {
            float a = hsel[i];
#pragma unroll
            for (int j = 0; j < i; ++j) a -= L[i * (i + 1) / 2 + j] * y[j];
            y[i] = a / L[i * (i + 1) / 2 + i];
        }
#pragma unroll
        for (int i = s; i >= 0; --i) {
            float a = y[i];
#pragma unroll
            for (int j = i + 1; j <= s; ++j) a -= L[j * (j + 1) / 2 + i] * cf[j];
            const float c = a / L[i * (i + 1) / 2 + i];
            cf[i] = __builtin_isfinite(c) ? c : 0.0f;
        }

        // ---- h = h_bar - coeffs . G[I]  (skip on last round) ----
        if (s < SPARSITY - 1) {
#pragma unroll
            for (int t = 0; t < 32; ++t) {
                float a = hb_r[t];
#pragma unroll
                for (int j = 0; j <= s; ++j)
                    a -= cf[j] * G[(size_t)I_[j] * NUM_ATOMS + t * 32 + lane];
                h[t] = __builtin_isfinite(a) ? a : 0.0f;
            }
        }
    }

    // ---- recon[n][d] = sum_s cf[s] * D[d][I[s]] ----
#pragma unroll
    for (int r = 0; r < 2; ++r) {
        const int dd = lane + r * 32;
        float a = 0.0f;
#pragma unroll
        for (int j = 0; j < SPARSITY; ++j)
            a += cf[j] * Dm[(size_t)dd * NUM_ATOMS + I_[j]];
        recon[(size_t)n * EMBED_DIM + dd] = a;
    }
    if (lane == 0) {
#pragma unroll
        for (int j = 0; j < SPARSITY; ++j) {
            outI[(size_t)n * SPARSITY + j] = (float)I_[j];
            outC[(size_t)n * SPARSITY + j] = cf[j];
        }
    }
}

extern "C" void kernel_launch(void* const* d_in, const int* in_sizes, int n_in,
                              void* d_out, int out_size, void* d_ws, size_t ws_size,
                              hipStream_t stream) {
    const float* X = (const float*)d_in[0];   // [64, 16384]
    const float* D = (const float*)d_in[1];   // [64, 1024]

    float* G    = (float*)d_ws;                               // 4 MB
    float* hbar = G + (size_t)NUM_ATOMS * NUM_ATOMS;          // 64 MB

    float* recon = (float*)d_out;                             // [16384, 64]
    float* outI  = recon + (size_t)N_PIXELS * EMBED_DIM;      // [16384, 8]
    float* outC  = outI + (size_t)N_PIXELS * SPARSITY;        // [16384, 8]

    // G = D^T D + eps*I : 64 M-tiles x 16 N-strips = 1024 waves
    wmma_gemm_tn<<<(NUM_ATOMS / 16) * (NUM_ATOMS / 64) / 4, 128, 0, stream>>>(
        D, NUM_ATOMS, D, NUM_ATOMS, G, NUM_ATOMS, DIAG_EPS);

    // h_bar = X^T D : 1024 M-tiles x 16 N-strips = 16384 waves
    wmma_gemm_tn<<<(N_PIXELS / 16) * (NUM_ATOMS / 64) / 4, 128, 0, stream>>>(
        X, N_PIXELS, D, NUM_ATOMS, hbar, NUM_ATOMS, 0.0f);

    // OMP: one wave per pixel, 8 pixels per 256-thread block
    omp_kernel<<<N_PIXELS / 8, 256, 0, stream>>>(G, hbar, D, recon, outI, outC);
}